// RoGPEMultiNetworkNodeEncoder_80281528697035
// MI455X (gfx1250) — compile-verified
//
#include <hip/hip_runtime.h>
#include <hip/hip_bf16.h>

// ---------------------------------------------------------------------------
// RoGPE node encoder for MI455X (gfx1250, wave32).
//  - MLP GEMMs use V_WMMA_F32_16X16X4_F32 (exact f32 matrix path).
//  - k-hop aggregation uses bit-packed boolean adjacency powers (A fits in L2).
// ---------------------------------------------------------------------------

typedef __attribute__((ext_vector_type(2))) float v2f;
typedef __attribute__((ext_vector_type(8))) float v8f;

#ifndef N_NODES_C
#define N_NODES_C 4096
#endif
#define DIM 128               // IN_DIM == hidden dim
#define WORDS_PER_ROW (N_NODES_C / 32)   // 128 u32 per adjacency row

// ---------------------------------------------------------------------------
// zero a range of 32-bit words
__global__ void zero_u32(unsigned int* __restrict__ p, int n) {
    int i = blockIdx.x * blockDim.x + threadIdx.x;
    if (i < n) p[i] = 0u;
}

// ---------------------------------------------------------------------------
// Y[M,N] = act( X[M,K] @ W[N,K]^T + bias ), via WMMA f32 16x16x4.
// One block = 16 output rows; 8 waves, one 16x16 N-tile per wave (N == 128).
__global__ __launch_bounds__(256) void gemm_relu_wmma(
    const float* __restrict__ X, const float* __restrict__ W,
    const float* __restrict__ bias, float* __restrict__ Y,
    int K, int N, int do_relu) {
    const int wave = threadIdx.x >> 5;    // 0..7 -> N tile
    const int lane = threadIdx.x & 31;
    const int half = lane >> 4;           // 0: K pair {0,1}, 1: {2,3}
    const int l    = lane & 15;
    const int m0   = blockIdx.x * 16;
    const int n0   = wave * 16;

    const float* __restrict__ xrow = X + (size_t)(m0 + l) * K;
    const float* __restrict__ wrow = W + (size_t)(n0 + l) * K;

    v8f acc = {};
    for (int k0 = 0; k0 < K; k0 += 4) {
        const int ka = k0 + (half << 1);
        v2f a = { xrow[ka], xrow[ka + 1] };   // A frag: 16x4, lane->M, half->K pair
        v2f b = { wrow[ka], wrow[ka + 1] };   // B frag: B[k][n] = W[n][k]
        acc = __builtin_amdgcn_wmma_f32_16x16x4_f32(
            false, a, false, b, (short)0, acc, false, false);
    }

    #pragma unroll
    for (int v = 0; v < 8; ++v) {
        const int m = m0 + v + 8 * half;      // C/D layout: VGPR v, half -> +8
        const int n = n0 + l;
        float val = acc[v];
        if (bias) val += bias[n];
        if (do_relu) val = fmaxf(val, 0.0f);
        Y[(size_t)m * N + n] = val;
    }
}

// ---------------------------------------------------------------------------
// a[node] = dot(H[node,:], W3[0,:]) + b3   (one wave per node)
__global__ __launch_bounds__(256) void head_matvec(
    const float* __restrict__ H, const float* __restrict__ W3,
    const float* __restrict__ b3, float* __restrict__ a, int K, int nNodes) {
    const int node = blockIdx.x * (blockDim.x >> 5) + (threadIdx.x >> 5);
    const int lane = threadIdx.x & 31;
    if (node >= nNodes) return;
    const float* __restrict__ h = H + (size_t)node * K;
    float s = 0.0f;
    for (int j = lane; j < K; j += 32) s += h[j] * W3[j];
    #pragma unroll
    for (int off = 16; off > 0; off >>= 1) s += __shfl_xor(s, off, 32);
    if (lane == 0) a[node] = s + b3[0];
}

// ---------------------------------------------------------------------------
// hop-1 edge scatter (duplicates kept) + bit-packed adjacency build (dedup)
__global__ void scatter_edges(const int* __restrict__ row,
                              const int* __restrict__ col,
                              const float* __restrict__ a0,
                              float* __restrict__ step,
                              unsigned int* __restrict__ Abits,
                              int nEdges) {
    int e = blockIdx.x * blockDim.x + threadIdx.x;
    if (e >= nEdges) return;
    const int r = row[e];
    const int c = col[e];
    atomicAdd(step + r, a0[c] * 0.6931471805599453f);   // ln(2)
    atomicOr(Abits + (size_t)r * WORDS_PER_ROW + (c >> 5), 1u << (c & 31));
}

// ---------------------------------------------------------------------------
// Boolean matmul on bitmatrices: Q[i,:] = OR_{k : P[i,k]} A[k,:]
// One block per row; scanned word is in LDS -> loop is block-uniform and the
// A-row OR loads are perfectly coalesced (A is 2MB, L2 resident).
__global__ __launch_bounds__(WORDS_PER_ROW) void bool_mm(
    const unsigned int* __restrict__ P, const unsigned int* __restrict__ A,
    unsigned int* __restrict__ Q) {
    __shared__ unsigned int srow[WORDS_PER_ROW];
    const int i = blockIdx.x;
    const int t = threadIdx.x;
    srow[t] = P[(size_t)i * WORDS_PER_ROW + t];
    __syncthreads();
    unsigned int acc = 0u;
    for (int w = 0; w < WORDS_PER_ROW; ++w) {
        unsigned int bits = srow[w];       // uniform across block
        while (bits) {
            const int b = __ffs(bits) - 1;
            bits &= bits - 1u;
            acc |= A[(size_t)(w * 32 + b) * WORDS_PER_ROW + t];
        }
    }
    Q[(size_t)i * WORDS_PER_ROW + t] = acc;
}

// ---------------------------------------------------------------------------
// rowsum[i] = sum_{j : Q[i,j]} a[j]   (one block per row, LDS reduction)
__global__ __launch_bounds__(WORDS_PER_ROW) void bitmask_matvec(
    const unsigned int* __restrict__ Q, const float* __restrict__ a,
    float* __restrict__ rowsum) {
    __shared__ float red[WORDS_PER_ROW];
    const int i = blockIdx.x;
    const int t = threadIdx.x;
    unsigned int bits = Q[(size_t)i * WORDS_PER_ROW + t];
    float s = 0.0f;
    while (bits) {
        const int b = __ffs(bits) - 1;
        bits &= bits - 1u;
        s += a[t * 32 + b];
    }
    red[t] = s;
    __syncthreads();
    #pragma unroll
    for (int off = WORDS_PER_ROW / 2; off > 0; off >>= 1) {
        if (t < off) red[t] += red[t + off];
        __syncthreads();
    }
    if (t == 0) rowsum[i] = red[0];
}

// ---------------------------------------------------------------------------
// step[i] += coeff * rowsum[i];  aout[i] = ain[i] + step[i]
// (hop 1: coeff = 0, rowsum = step -> pure "a += step")
__global__ void update_step_a(float* __restrict__ step,
                              const float* __restrict__ rowsum, float coeff,
                              const float* __restrict__ ain,
                              float* __restrict__ aout, int n) {
    int i = blockIdx.x * blockDim.x + threadIdx.x;
    if (i >= n) return;
    const float s = step[i] + coeff * rowsum[i];
    step[i] = s;
    aout[i] = ain[i] + s;
}

// ---------------------------------------------------------------------------
extern "C" void kernel_launch(void* const* d_in, const int* in_sizes, int n_in,
                              void* d_out, int out_size, void* d_ws, size_t ws_size,
                              hipStream_t stream) {
    (void)n_in; (void)ws_size;
    const float* coeffs = (const float*)d_in[0];
    const int*   edges  = (const int*)d_in[1];   // [2, nEdges] int32
    const float* W0     = (const float*)d_in[2];
    const float* b0     = (const float*)d_in[3];
    const float* W1     = (const float*)d_in[4];
    const float* W2     = (const float*)d_in[5];
    const float* W3     = (const float*)d_in[6];
    const float* b3     = (const float*)d_in[7];

    const int nNodes = in_sizes[0] / DIM;        // 4096
    const int nEdges = in_sizes[1] / 2;          // 131072

    // ---- workspace partition ----
    float* h1   = (float*)d_ws;                          // [nNodes, DIM]
    float* h2   = h1 + (size_t)nNodes * DIM;             // [nNodes, DIM]
    float* avec = h2 + (size_t)nNodes * DIM;             // [nNodes]
    float* step = avec + nNodes;                         // [nNodes]
    float* rsum = step + nNodes;                         // [nNodes]
    unsigned int* Abits = (unsigned int*)(rsum + nNodes);         // [nNodes,128]
    unsigned int* P2    = Abits + (size_t)nNodes * WORDS_PER_ROW; // A^2
    unsigned int* P3    = P2    + (size_t)nNodes * WORDS_PER_ROW; // A^3

    // ---- 0) zero step/rowsum/Abits (contiguous region) ----
    {
        const int nz = 2 * nNodes + nNodes * WORDS_PER_ROW;
        zero_u32<<<(nz + 255) / 256, 256, 0, stream>>>((unsigned int*)step, nz);
    }

    // ---- 1) MLP via f32 WMMA ----
    gemm_relu_wmma<<<nNodes / 16, 256, 0, stream>>>(coeffs, W0, b0,      h1, DIM, DIM, 1);
    gemm_relu_wmma<<<nNodes / 16, 256, 0, stream>>>(h1,     W1, nullptr, h2, DIM, DIM, 1);
    gemm_relu_wmma<<<nNodes / 16, 256, 0, stream>>>(h2,     W2, nullptr, h1, DIM, DIM, 1);
    head_matvec<<<(nNodes * 32 + 255) / 256, 256, 0, stream>>>(h1, W3, b3, avec, DIM, nNodes);

    // ---- 2) hop 1: raw edge scatter (ln2) + adjacency bitmatrix ----
    scatter_edges<<<(nEdges + 255) / 256, 256, 0, stream>>>(
        edges, edges + nEdges, avec, step, Abits, nEdges);
    update_step_a<<<(nNodes + 255) / 256, 256, 0, stream>>>(
        step, step, 0.0f, avec, avec, nNodes);                    // a1 = a0 + step

    // ---- 3) hop 2: A2 = A & A (boolean); step += ln3 * (A2 @ a1) ----
    bool_mm<<<nNodes, WORDS_PER_ROW, 0, stream>>>(Abits, Abits, P2);
    bitmask_matvec<<<nNodes, WORDS_PER_ROW, 0, stream>>>(P2, avec, rsum);
    update_step_a<<<(nNodes + 255) / 256, 256, 0, stream>>>(
        step, rsum, 1.0986122886681098f /*ln3*/, avec, avec, nNodes);  // a2

    // ---- 4) hop 3: A3 = A2 & A; step += ln4 * (A3 @ a2); out = a2 + step ----
    bool_mm<<<nNodes, WORDS_PER_ROW, 0, stream>>>(P2, Abits, P3);
    bitmask_matvec<<<nNodes, WORDS_PER_ROW, 0, stream>>>(P3, avec, rsum);
    update_step_a<<<(nNodes + 255) / 256, 256, 0, stream>>>(
        step, rsum, 1.3862943611198906f /*ln4*/, avec, (float*)d_out, nNodes);

    (void)out_size;
}